// MSAColumnGlobalAttention_52501680226864
// MI455X (gfx1250) — compile-verified
//
#include <hip/hip_runtime.h>
#include <math.h>
#include <stdint.h>

#define SDIM 2048
#define IDIM 384
#define CDIM 64
#define NW   8
#define TPB  256
#define NTILES 128        // SDIM/16
#define JTILES (NTILES / NW)
#define ROWP 68           // padded row stride (floats) == TDM pad 4dw / 64dw

typedef float v2f __attribute__((ext_vector_type(2)));
typedef float v8f __attribute__((ext_vector_type(8)));
typedef unsigned int u32x4 __attribute__((ext_vector_type(4)));
typedef int i32x8 __attribute__((ext_vector_type(8)));
typedef int i32x4 __attribute__((ext_vector_type(4)));

// D = A(16x4) * B(4x16) + C, fp32 WMMA (wave32):
//   A: lane m=lane&15, kbase=(lane&16)?2:0 -> {a[m][kb], a[m][kb+1]}
//   B: lane n=lane&15, kbase=(lane&16)?2:0 -> {b[kb][n], b[kb+1][n]}
//   C/D: vgpr j -> row (j + (lane<16?0:8)), col lane&15
static __device__ __forceinline__ v8f wmma_f32(v2f a, v2f b, v8f c) {
  return __builtin_amdgcn_wmma_f32_16x16x4_f32(false, a, false, b, (short)0, c,
                                               false, false);
}

// TDM: fetch a 16-row x 64-float tile of m (row stride IDIM*CDIM elements)
// into LDS at lds_off, hardware-padding +4 DWORDs after every 64 DWORDs so
// rows land at ROWP=68-float stride (conflict-free WMMA A-fragment reads).
static __device__ __forceinline__ void tdm_load_tile(const float* gptr,
                                                     uint32_t lds_off) {
  uint64_t ga = (uint64_t)(uintptr_t)gptr;
  uint32_t ga_lo = (uint32_t)__builtin_amdgcn_readfirstlane((int)(uint32_t)ga);
  uint32_t ga_hi = (uint32_t)__builtin_amdgcn_readfirstlane((int)(uint32_t)(ga >> 32));
  uint32_t lo = (uint32_t)__builtin_amdgcn_readfirstlane((int)lds_off);
  u32x4 g0;
  g0[0] = 1u;                                    // count=1, no gather
  g0[1] = lo;                                    // lds_addr (bytes)
  g0[2] = ga_lo;                                 // global_addr[31:0]
  g0[3] = (ga_hi & 0x01FFFFFFu) | (2u << 30);    // global_addr[56:32] | type=2
  i32x8 g1;
  g1[0] = 0x07520000;        // data_size=4B | pad_enable | interval=64dw | amount=4dw
  g1[1] = 64 << 16;          // tensor_dim0 = 64 (lo16 in [31:16])
  g1[2] = 16 << 16;          // tensor_dim1 = 16
  g1[3] = 64 << 16;          // tile_dim0 = 64
  g1[4] = 16;                // tile_dim1 = 16, tile_dim2 = 0
  g1[5] = IDIM * CDIM;       // tensor_dim0_stride = 24576 elements
  g1[6] = 0;
  g1[7] = 0;
  i32x4 z4 = {};
  i32x8 z8 = {};
  __builtin_amdgcn_tensor_load_to_lds(g0, g1, z4, z4, z8, 0);
}

static __device__ __forceinline__ void cfence() {
  asm volatile("" ::: "memory");  // compiler-only fence; DS ops are in-order per wave
}

__global__ __launch_bounds__(TPB) void msa_col_global_attn(
    const float* __restrict__ m, const float* __restrict__ mask,
    const float* __restrict__ gam, const float* __restrict__ bet,
    const float* __restrict__ Wq, const float* __restrict__ Wk,
    const float* __restrict__ Wv, const float* __restrict__ Wg,
    const float* __restrict__ bg, const float* __restrict__ Wo,
    const float* __restrict__ bo, float* __restrict__ out) {
  __shared__ float kvS[SDIM * 16];       // 128 KB: k (cols 0-7) | v (cols 8-15)
  __shared__ float meta[SDIM * 4];       // 32 KB: mu, rstd, mask per row
  __shared__ v2f   WkvB[16 * 32];        // 4 KB  B-fragments, fused k|v GEMM
  __shared__ v2f   WgB[4 * 16 * 32];     // 16 KB B-fragments, gate GEMM
  __shared__ v2f   WoB[4 * 16 * 32];     // 16 KB B-fragments, out-proj GEMM
  __shared__ __align__(16) float mst[NW * 2 * 16 * ROWP];  // 68 KB dbl-buf tiles
  __shared__ float gamS[64], betS[64], bgS[64], boS[64];
  __shared__ float qgS[65];              // q_global numerators + denom
  __shared__ float qS[64], ctxS[64];

  const int i    = blockIdx.x;     // column index (I dimension)
  const int tid  = threadIdx.x;
  const int wave = tid >> 5;
  const int lane = tid & 31;
  const int nl   = lane & 15;
  const int kb   = (lane & 16) ? 2 : 0;

  // ---------- phase 0: stage weights as WMMA B-fragments ----------
  for (int e = tid; e < 16 * 32; e += TPB) {
    int t = e >> 5, ln = e & 31;
    int k0 = 4 * t + ((ln & 16) ? 2 : 0);
    int n = ln & 15;
    v2f w;
    if (n < 8) { w.x = Wk[n * CDIM + k0];       w.y = Wk[n * CDIM + k0 + 1]; }
    else       { w.x = Wv[(n - 8) * CDIM + k0]; w.y = Wv[(n - 8) * CDIM + k0 + 1]; }
    WkvB[e] = w;
  }
  for (int e = tid; e < 4 * 16 * 32; e += TPB) {
    int nt = e >> 9, t = (e >> 5) & 15, ln = e & 31;
    int k0 = 4 * t + ((ln & 16) ? 2 : 0);
    int n = nt * 16 + (ln & 15);
    v2f wg; wg.x = Wg[n * CDIM + k0]; wg.y = Wg[n * CDIM + k0 + 1];
    WgB[e] = wg;
    v2f wo; wo.x = Wo[n * CDIM + k0]; wo.y = Wo[n * CDIM + k0 + 1];
    WoB[e] = wo;
  }
  if (tid < 64) { gamS[tid] = gam[tid]; betS[tid] = bet[tid];
                  bgS[tid]  = bg[tid];  boS[tid]  = bo[tid]; }
  if (tid < 65) qgS[tid] = 0.f;

  // per-lane normalize mapping: row = lane>>1, colbase = (lane&1)*32
  const int nrow = lane >> 1;
  const int cb   = (lane & 1) * 32;

  float* const mw0 = &mst[wave * 2 * 16 * ROWP];
  float* const mw1 = mw0 + 16 * ROWP;
  const uint32_t mw_off0 = (uint32_t)(uintptr_t)mw0;  // LDS byte offset
  const uint32_t mw_off1 = (uint32_t)(uintptr_t)mw1;

  // prefetch first tile of phase 1 (overlaps with weight staging)
  tdm_load_tile(&m[((size_t)(wave * 16) * IDIM + i) * CDIM], mw_off0);

  __syncthreads();   // weights/LN params visible

  float gr[32], br[32];
#pragma unroll
  for (int e = 0; e < 32; ++e) { gr[e] = gamS[cb + e]; br[e] = betS[cb + e]; }

  float qgacc[32];
#pragma unroll
  for (int e = 0; e < 32; ++e) qgacc[e] = 0.f;
  float dacc = 0.f;

  // ---------- phase 1: LN stats + normalize + q_global accum + k|v WMMA ----
  // All tile state (mst buffer, meta rows, kvS rows) is wave-private; no
  // block barriers needed inside the loop -> waves slip to hide TDM latency.
  for (int j = 0; j < JTILES; ++j) {
    const int tile = wave + j * NW;
    const int tb = tile * 16;
    float* mw = (j & 1) ? mw1 : mw0;
    if (j + 1 < JTILES) {  // prefetch next tile into other buffer
      tdm_load_tile(&m[((size_t)(tb + NW * 16) * IDIM + i) * CDIM],
                    (j & 1) ? mw_off0 : mw_off1);
      __builtin_amdgcn_s_wait_tensorcnt(1);   // current tile landed
    } else {
      __builtin_amdgcn_s_wait_tensorcnt(0);
    }
    cfence();
    {                                        // mean / var per row
      int r = nl, qh = (lane >> 4) * 8;
      float s1 = 0.f, s2 = 0.f;
#pragma unroll
      for (int q = 0; q < 8; ++q) {
        float4 x = *(const float4*)&mw[r * ROWP + (qh + q) * 4];
        s1 += x.x + x.y + x.z + x.w;
        s2 += x.x * x.x + x.y * x.y + x.z * x.z + x.w * x.w;
      }
      s1 += __shfl_xor(s1, 16, 32);
      s2 += __shfl_xor(s2, 16, 32);
      float mu   = s1 * (1.f / 64.f);
      float var  = s2 * (1.f / 64.f) - mu * mu;
      float rstd = rsqrtf(var + 1e-5f);
      if (lane < 16) {
        float mk = mask[(size_t)(tb + r) * IDIM + i];
        meta[(tb + r) * 4 + 0] = mu;
        meta[(tb + r) * 4 + 1] = rstd;
        meta[(tb + r) * 4 + 2] = mk;
      }
    }
    cfence();
    {                                        // normalize in LDS + masked accum
      float mu   = meta[(tb + nrow) * 4 + 0];
      float rstd = meta[(tb + nrow) * 4 + 1];
      float mk   = meta[(tb + nrow) * 4 + 2];
#pragma unroll
      for (int e = 0; e < 32; ++e) {
        float x  = mw[nrow * ROWP + cb + e];
        float xn = (x - mu) * rstd * gr[e] + br[e];
        mw[nrow * ROWP + cb + e] = xn;
        qgacc[e] += mk * xn;
      }
      if ((lane & 1) == 0) dacc += mk;
    }
    cfence();
    {                                        // fused k|v GEMM: [16x64]@[64x16]
      v8f acc = {};
#pragma unroll
      for (int t = 0; t < 16; ++t) {
        v2f a = *(const v2f*)&mw[nl * ROWP + 4 * t + kb];
        v2f b = WkvB[t * 32 + lane];
        acc = wmma_f32(a, b, acc);
      }
      int rbase = tb + ((lane & 16) ? 8 : 0);
#pragma unroll
      for (int jj = 0; jj < 8; ++jj) kvS[(rbase + jj) * 16 + nl] = acc[jj];
    }
    cfence();
  }
  // prefetch phase-4 first tile now; TDM runs underneath phases 2-3
  tdm_load_tile(&m[((size_t)(wave * 16) * IDIM + i) * CDIM], mw_off0);

#pragma unroll
  for (int e = 0; e < 32; ++e) atomicAdd(&qgS[cb + e], qgacc[e]);
  if ((lane & 1) == 0) atomicAdd(&qgS[64], dacc);
  __syncthreads();

  // ---------- phase 2: q = (q_global/denom) @ Wq^T, scale folded in --------
  if (tid < 64) {
    float inv = 1.f / fmaxf(qgS[64], 1e-5f);
    float qv = 0.f;
#pragma unroll 8
    for (int c = 0; c < 64; ++c) qv += (qgS[c] * inv) * Wq[tid * CDIM + c];
    qS[tid] = qv * 0.35355339059327373f;  // 1/sqrt(CH)
  }
  __syncthreads();

  // ---------- phase 3: online-softmax attention, one head per wave ---------
  {
    const int h = wave;
    float qh[8];
#pragma unroll
    for (int ch = 0; ch < 8; ++ch) qh[ch] = qS[h * 8 + ch];
    float mrun = -INFINITY, srun = 0.f;
    float crun[8];
#pragma unroll
    for (int ch = 0; ch < 8; ++ch) crun[ch] = 0.f;
    for (int sb = 0; sb < SDIM / 32; ++sb) {
      int s = sb * 32 + lane;
      const float4 k0 = *(const float4*)&kvS[s * 16 + 0];
      const float4 k1 = *(const float4*)&kvS[s * 16 + 4];
      float l = qh[0] * k0.x + qh[1] * k0.y + qh[2] * k0.z + qh[3] * k0.w +
                qh[4] * k1.x + qh[5] * k1.y + qh[6] * k1.z + qh[7] * k1.w;
      float mk = meta[s * 4 + 2];
      l = (mk > 0.f) ? l : -1e9f;
      float nm   = fmaxf(mrun, l);
      float p    = __expf(l - nm);
      float corr = __expf(mrun - nm);
      const float4 v0 = *(const float4*)&kvS[s * 16 + 8];
      const float4 v1 = *(const float4*)&kvS[s * 16 + 12];
      srun = srun * corr + p;
      crun[0] = crun[0] * corr + p * v0.x;  crun[1] = crun[1] * corr + p * v0.y;
      crun[2] = crun[2] * corr + p * v0.z;  crun[3] = crun[3] * corr + p * v0.w;
      crun[4] = crun[4] * corr + p * v1.x;  crun[5] = crun[5] * corr + p * v1.y;
      crun[6] = crun[6] * corr + p * v1.z;  crun[7] = crun[7] * corr + p * v1.w;
      mrun = nm;
    }
#pragma unroll
    for (int off = 16; off > 0; off >>= 1) {   // butterfly flash-combine
      float om = __shfl_xor(mrun, off, 32);
      float os = __shfl_xor(srun, off, 32);
      float oc[8];
#pragma unroll
      for (int ch = 0; ch < 8; ++ch) oc[ch] = __shfl_xor(crun[ch], off, 32);
      float nm = fmaxf(mrun, om);
      float c1 = __expf(mrun - nm), c2 = __expf(om - nm);
      srun = srun * c1 + os * c2;
#pragma unroll
      for (int ch = 0; ch < 8; ++ch) crun[ch] = crun[ch] * c1 + oc[ch] * c2;
      mrun = nm;
    }
    if (lane == 0) {
      float inv = 1.f / srun;
#pragma unroll
      for (int ch = 0; ch < 8; ++ch) ctxS[h * 8 + ch] = crun[ch] * inv;
    }
  }
  __syncthreads();

  // ---------- phase 4: gate GEMM -> sigmoid*ctx -> out-proj GEMM -----------
  for (int j = 0; j < JTILES; ++j) {
    const int tile = wave + j * NW;
    const int tb = tile * 16;
    float* mw = (j & 1) ? mw1 : mw0;
    if (j + 1 < JTILES) {
      tdm_load_tile(&m[((size_t)(tb + NW * 16) * IDIM + i) * CDIM],
                    (j & 1) ? mw_off0 : mw_off1);
      __builtin_amdgcn_s_wait_tensorcnt(1);
    } else {
      __builtin_amdgcn_s_wait_tensorcnt(0);
    }
    cfence();
    {                                        // normalize with cached stats
      float mu   = meta[(tb + nrow) * 4 + 0];
      float rstd = meta[(tb + nrow) * 4 + 1];
#pragma unroll
      for (int e = 0; e < 32; ++e) {
        float x = mw[nrow * ROWP + cb + e];
        mw[nrow * ROWP + cb + e] = (x - mu) * rstd * gr[e] + br[e];
      }
    }
    cfence();
    v8f og[4];
#pragma unroll
    for (int nt = 0; nt < 4; ++nt) {         // GEMM1: m_norm @ Wg^T
      v8f acc = {};
#pragma unroll
      for (int t = 0; t < 16; ++t) {
        v2f a = *(const v2f*)&mw[nl * ROWP + 4 * t + kb];
        v2f b = WgB[(nt * 16 + t) * 32 + lane];
        acc = wmma_f32(a, b, acc);
      }
      int n = nt * 16 + nl;
      float bgv = bgS[n], ctxv = ctxS[n];
#pragma unroll
      for (int jj = 0; jj < 8; ++jj) {       // o = ctx * sigmoid(.)
        float gv = 1.f / (1.f + __expf(-(acc[jj] + bgv)));
        og[nt][jj] = ctxv * gv;
      }
    }
    cfence();
    {                                        // transpose o frags via LDS
      int r0 = (lane & 16) ? 8 : 0;
#pragma unroll
      for (int nt = 0; nt < 4; ++nt)
#pragma unroll
        for (int jj = 0; jj < 8; ++jj)
          mw[(r0 + jj) * ROWP + nt * 16 + nl] = og[nt][jj];
    }
    cfence();
#pragma unroll
    for (int nt = 0; nt < 4; ++nt) {         // GEMM2: o @ Wo^T + bo -> out
      v8f acc = {};
#pragma unroll
      for (int t = 0; t < 16; ++t) {
        v2f a = *(const v2f*)&mw[nl * ROWP + 4 * t + kb];
        v2f b = WoB[(nt * 16 + t) * 32 + lane];
        acc = wmma_f32(a, b, acc);
      }
      float bov = boS[nt * 16 + nl];
      int r0 = tb + ((lane & 16) ? 8 : 0);
#pragma unroll
      for (int jj = 0; jj < 8; ++jj)
        out[((size_t)(r0 + jj) * IDIM + i) * CDIM + nt * 16 + nl] = acc[jj] + bov;
    }
    cfence();
  }
}

extern "C" void kernel_launch(void* const* d_in, const int* in_sizes, int n_in,
                              void* d_out, int out_size, void* d_ws,
                              size_t ws_size, hipStream_t stream) {
  (void)in_sizes; (void)n_in; (void)out_size; (void)d_ws; (void)ws_size;
  const float* m    = (const float*)d_in[0];
  const float* mask = (const float*)d_in[1];
  const float* gam  = (const float*)d_in[2];
  const float* bet  = (const float*)d_in[3];
  const float* Wq   = (const float*)d_in[4];
  const float* Wk   = (const float*)d_in[5];
  const float* Wv   = (const float*)d_in[6];
  const float* Wg   = (const float*)d_in[7];
  const float* bg   = (const float*)d_in[8];
  const float* Wo   = (const float*)d_in[9];
  const float* bo   = (const float*)d_in[10];
  float* out = (float*)d_out;
  hipLaunchKernelGGL(msa_col_global_attn, dim3(IDIM), dim3(TPB), 0, stream,
                     m, mask, gam, bet, Wq, Wk, Wv, Wg, bg, Wo, bo, out);
}